// PositionalEmbedding_51342039056881
// MI455X (gfx1250) — compile-verified
//
#include <hip/hip_runtime.h>
#include <stdint.h>

// out[n, :] = W[x[n], :] * (x[n] != 0) + PE(n+1, :)
// N = 131072 rows, DIM = 1024 fp32 per row.
// Memory-bound: ~0.5 GB NT writes + L2-resident gather of a ~206 MB table
// (192 MB L2). Use gfx1250 async global->LDS staging (ASYNCcnt pipeline)
// + non-temporal stores so the output stream does not evict W from L2.

#define TPB    256          // 8 wave32 waves
#define DIM    1024         // floats per row; TPB * 4 floats = 1024
#define ROWS   16           // rows per block
#define STAGES 4            // async pipeline depth (STAGES-1 = 3 in flight)

typedef float v4f __attribute__((ext_vector_type(4)));
typedef __attribute__((address_space(3))) unsigned char lds_byte_t;

// generic pointer -> 32-bit LDS byte address
__device__ __forceinline__ uint32_t lds_addr_of(const void* p) {
    return (uint32_t)(uintptr_t)(lds_byte_t*)p;
}

// gfx1250 async DMA: copies 16 bytes from global address gbase + voff_bytes to LDS address lds_dst
// one instruction per wave -> one ASYNCcnt token per wave
__device__ __forceinline__ void async_copy_b128(uint32_t lds_dst, uint64_t gbase,
                                                uint32_t voff_bytes) {
    asm volatile("global_load_async_to_lds_b128 %0, %1, %2"
                 :
                 : "v"(lds_dst), "v"(voff_bytes), "s"(gbase)
                 : "memory");
}

template <int N>
__device__ __forceinline__ void wait_asynccnt() {
    asm volatile("s_wait_asynccnt %0" ::"n"(N) : "memory");
}

// explicit LDS read so the compiler cannot reason away data written by the
// async DMA (which it cannot see as an LDS store)
__device__ __forceinline__ v4f lds_read_b128(uint32_t addr) {
    v4f r;
    asm volatile("ds_load_b128 %0, %1\n\t"
                 "s_wait_dscnt 0x0"
                 : "=v"(r)
                 : "v"(addr)
                 : "memory");
    return r;
}

__global__ __launch_bounds__(TPB) void pe_embed_kernel(
    const int* __restrict__ x, const float* __restrict__ W,
    float* __restrict__ out, int N) {
    __shared__ v4f sbuf[STAGES][TPB];

    const int tid  = threadIdx.x;
    const int row0 = blockIdx.x * ROWS;
    const int Nm1  = N - 1;
    const uint32_t voff = (uint32_t)(tid * 16);   // byte offset within a 4KB row

    uint32_t lds_a[STAGES];
#pragma unroll
    for (int s = 0; s < STAGES; ++s) lds_a[s] = lds_addr_of(&sbuf[s][tid]);

    // Per-thread PE frequencies: this thread owns dims j0..j0+3.
    //   even j (cos): w = 10000^(-j/D);  odd j (sin): w = 10000^(-(j+1)/D)
    // -> only 3 unique exponents per thread: j0, j0+2, j0+4
    const float j0 = (float)(tid * 4);
    const float k  = 13.287712379549449f / (float)DIM;     // log2(10000)/D
    const float wa = __builtin_amdgcn_exp2f(-j0 * k);           // cos @ j0
    const float wb = __builtin_amdgcn_exp2f(-(j0 + 2.0f) * k);  // sin @ j0+1, cos @ j0+2
    const float wc = __builtin_amdgcn_exp2f(-(j0 + 4.0f) * k);  // sin @ j0+3

    // ---- prologue: fill the async pipeline (STAGES-1 rows in flight) ----
#pragma unroll
    for (int p = 0; p < STAGES - 1; ++p) {
        int rr = row0 + p;
        rr = rr < Nm1 ? rr : Nm1;
        const int idx = x[rr];                    // uniform -> s_load (KMcnt)
        async_copy_b128(lds_a[p],
                        (uint64_t)(const void*)(W + (size_t)idx * DIM), voff);
    }

    // ---- steady state: issue r+3, wait for r, consume r ----
#pragma unroll
    for (int r = 0; r < ROWS; ++r) {
        if (r + STAGES - 1 < ROWS) {
            int rr = row0 + r + STAGES - 1;
            rr = rr < Nm1 ? rr : Nm1;
            const int idx = x[rr];
            async_copy_b128(lds_a[(r + STAGES - 1) % STAGES],
                            (uint64_t)(const void*)(W + (size_t)idx * DIM), voff);
            wait_asynccnt<STAGES - 1>();          // oldest (row r) complete
        } else {
            wait_asynccnt<0>();                   // pipeline drain (last 3 rows)
        }

        const int row  = row0 + r;
        const int rowc = row < Nm1 ? row : Nm1;
        const v4f e    = lds_read_b128(lds_a[r % STAGES]);

        const float scale = (x[rowc] != 0) ? 1.0f : 0.0f;   // padding_idx row = 0
        const float pos   = (float)(rowc + 1);              // 1-based position

        v4f o;
        o.x = fmaf(e.x, scale, __cosf(pos * wa));
        o.y = fmaf(e.y, scale, __sinf(pos * wb));
        o.z = fmaf(e.z, scale, __cosf(pos * wb));
        o.w = fmaf(e.w, scale, __sinf(pos * wc));

        if (row < N) {
            // NT store: keep the 512MB output stream out of L2 so the ~206MB
            // embedding table stays resident in the 192MB L2
            __builtin_nontemporal_store(
                o, (v4f*)(out + (size_t)row * DIM + (size_t)tid * 4));
        }
    }
}

extern "C" void kernel_launch(void* const* d_in, const int* in_sizes, int n_in,
                              void* d_out, int out_size, void* d_ws, size_t ws_size,
                              hipStream_t stream) {
    (void)n_in; (void)out_size; (void)d_ws; (void)ws_size;
    const int*   x = (const int*)d_in[0];
    const float* W = (const float*)d_in[1];
    float*     out = (float*)d_out;
    const int    N = in_sizes[0];                 // 131072
    const int blocks = (N + ROWS - 1) / ROWS;     // 8192 blocks x 256 threads
    pe_embed_kernel<<<blocks, TPB, 0, stream>>>(x, W, out, N);
}